// CNN_56745107915038
// MI455X (gfx1250) — compile-verified
//
#include <hip/hip_runtime.h>
#include <math.h>

typedef __attribute__((ext_vector_type(2))) float v2f;
typedef __attribute__((ext_vector_type(8))) float v8f;

#define EEG_CH 64
#define WIN    128
#define CWN    (10 * 64 * 9)   // 5760 conv weights

// One fused workgroup: 256 threads = 8 wave32s on one WGP.
__global__ __launch_bounds__(256)
void eegnet_fused(const float* __restrict__ xg,       // (1,1,66,128)
                  const float* __restrict__ se_w1,    // (64,64)
                  const float* __restrict__ se_b1,    // (64)
                  const float* __restrict__ se_w2,    // (64,64)
                  const float* __restrict__ se_b2,    // (64)
                  const float* __restrict__ convw,    // (10,1,64,9)
                  const float* __restrict__ convb,    // (10)
                  const float* __restrict__ fcn_w1,   // (10,20)
                  const float* __restrict__ fcn_b1,   // (10)
                  const float* __restrict__ fcn_w2,   // (2,10)
                  const float* __restrict__ fcn_b2,   // (2)
                  float* __restrict__ out)            // (1,2)
{
    __shared__ float eeg_lds[EEG_CH * WIN];   // 32 KB, channels 1..64 of x
    __shared__ float cw_lds[CWN];             // 22.5 KB, conv weights
    __shared__ float ss[EEG_CH];              // per-channel sum of squares
    __shared__ float wstats[4];               // dotA, dotB, ssA, ssB
    __shared__ float eegr[2 * 64];
    __shared__ float hbuf[2 * 64];
    __shared__ float sbuf[2 * 64];
    __shared__ float denom[2];
    __shared__ float scale_lds[2 * 64];       // SE softmax weights
    __shared__ float acc[20];                 // relu-sum accumulators (o*2+h)

    const int tid  = threadIdx.x;
    const int lane = tid & 31;
    const int wave = tid >> 5;
    const int half = lane >> 4;               // 0: lanes 0-15, 1: lanes 16-31
    const int lm   = lane & 15;               // M for A-frags, N for B-frags
    const int j0   = 2 * half;                // K sub-offset per ISA layout

    // ---------------- Phase 0: stage eeg + conv_w into LDS ---------------
    for (int e = tid; e < EEG_CH * WIN; e += 256) eeg_lds[e] = xg[WIN + e];
    for (int e = tid; e < CWN; e += 256)         cw_lds[e]  = convw[e];
    if (tid < 20) acc[tid] = 0.f;
    __syncthreads();

    // ---------------- Phase 1: norms + the diagonal-quirk dot products ---
    if (tid < 64) {
        float s = 0.f;
        const float* row = eeg_lds + tid * WIN;
        for (int w = 0; w < WIN; ++w) s += row[w] * row[w];
        ss[tid] = s;
    } else if (tid == 64) {                    // ||wav_a||^2
        float s = 0.f;
        for (int w = 0; w < WIN; ++w) { float v = xg[w]; s += v * v; }
        wstats[2] = s;
    } else if (tid == 65) {                    // ||wav_b||^2
        float s = 0.f;
        for (int w = 0; w < WIN; ++w) { float v = xg[65 * WIN + w]; s += v * v; }
        wstats[3] = s;
    } else if (tid == 66) {                    // dot(ch0, wav_a)  (diag quirk)
        float s = 0.f;
        for (int w = 0; w < WIN; ++w) s += eeg_lds[w] * xg[w];
        wstats[0] = s;
    } else if (tid == 67) {                    // dot(ch0, wav_b)
        float s = 0.f;
        for (int w = 0; w < WIN; ++w) s += eeg_lds[w] * xg[65 * WIN + w];
        wstats[1] = s;
    }
    __syncthreads();

    // ---------------- Phase 2: eeg_r[r][c] = dot_r / (||ch_c|| * ||wav_r||)
    if (tid < 128) {
        const int r = tid >> 6, c = tid & 63;
        eegr[tid] = wstats[r] / (sqrtf(ss[c]) * sqrtf(wstats[2 + r]));
    }
    __syncthreads();

    // ---------------- Phase 3: h = tanh(eeg_r @ se_w1^T + se_b1) ---------
    if (tid < 128) {
        const int r = tid >> 6, c = tid & 63;
        float s = se_b1[c];
        const float* er = eegr + r * 64;
        const float* wr = se_w1 + c * 64;
        for (int k = 0; k < 64; ++k) s += er[k] * wr[k];
        hbuf[tid] = tanhf(s);
    }
    __syncthreads();

    // ---------------- Phase 4: s = sigmoid(h @ se_w2^T + se_b2) ----------
    if (tid < 128) {
        const int r = tid >> 6, c = tid & 63;
        float s = se_b2[c];
        const float* hr = hbuf + r * 64;
        const float* wr = se_w2 + c * 64;
        for (int k = 0; k < 64; ++k) s += hr[k] * wr[k];
        sbuf[tid] = 1.f / (1.f + expf(-s));
    }
    __syncthreads();

    // ---------------- Phase 5/6: row softmax -> SE scale -----------------
    if (tid < 2) {
        float s = 0.f;
        for (int c = 0; c < 64; ++c) s += expf(sbuf[tid * 64 + c]);
        denom[tid] = s;
    }
    __syncthreads();
    if (tid < 128) scale_lds[tid] = expf(sbuf[tid]) / denom[tid >> 6];
    __syncthreads();

    // ---------------- Phase 7: conv as WMMA f32 16x16x4 GEMM -------------
    // A = conv_w (16x576, rows 10..15 masked to 0); B built on the fly from
    // LDS eeg * SE scale. 16 h-pure N-tiles of 16 cols; 2 tiles per wave.
    // Branch-free inner loop: clamped LDS addresses + 0/1 mask multiplies.
    const float mm   = (lm < 10) ? 1.f : 0.f;   // pad M rows 10..15
    const int   mrow = (lm < 10) ? lm : 0;      // clamped A row
    const float hm   = (half == 0) ? 1.f : 0.f; // K-pad tail (only k=8 valid)

    for (int t = wave; t < 16; t += 8) {
        const int hh = t >> 3;                  // conv output row (0/1)
        const int colbase = (t & 7) * 16;       // width base (valid w < 120)
        const int w0 = colbase + lm;            // column N = lane&15
        v8f c = {0.f, 0.f, 0.f, 0.f, 0.f, 0.f, 0.f, 0.f};

        for (int i = 0; i < EEG_CH; ++i) {
            const float sc = scale_lds[hh * 64 + i];
            const int wb = mrow * 576 + i * 9;  // A row base in cw_lds
            const int eb = i * WIN;             // eeg row base
            v2f a, b;

            // K-chunk 0: k = j0, j0+1  (0..3, always valid)
            a[0] = cw_lds[wb + j0] * mm;
            a[1] = cw_lds[wb + j0 + 1] * mm;
            {
                int i0 = w0 + j0;     if (i0 > 127) i0 = 127;
                int i1 = w0 + j0 + 1; if (i1 > 127) i1 = 127;
                b[0] = eeg_lds[eb + i0] * sc;
                b[1] = eeg_lds[eb + i1] * sc;
            }
            c = __builtin_amdgcn_wmma_f32_16x16x4_f32(
                    false, a, false, b, (short)0, c, false, false);

            // K-chunk 1: k = 4+j0, 5+j0  (4..7, always valid)
            a[0] = cw_lds[wb + 4 + j0] * mm;
            a[1] = cw_lds[wb + 5 + j0] * mm;
            {
                int i0 = w0 + 4 + j0; if (i0 > 127) i0 = 127;
                int i1 = w0 + 5 + j0; if (i1 > 127) i1 = 127;
                b[0] = eeg_lds[eb + i0] * sc;
                b[1] = eeg_lds[eb + i1] * sc;
            }
            c = __builtin_amdgcn_wmma_f32_16x16x4_f32(
                    false, a, false, b, (short)0, c, false, false);

            // K-chunk 2 (pad 9->12): only k=8 contributes, lanes 0-15 only.
            a[0] = cw_lds[wb + 8] * (mm * hm);
            a[1] = 0.f;
            {
                int i0 = w0 + 8; if (i0 > 127) i0 = 127;
                b[0] = eeg_lds[eb + i0] * (sc * hm);
                b[1] = 0.f;
            }
            c = __builtin_amdgcn_wmma_f32_16x16x4_f32(
                    false, a, false, b, (short)0, c, false, false);
        }

        // ReLU(y + b) and accumulate the width-mean numerator.
#pragma unroll
        for (int r = 0; r < 8; ++r) {
            const int o = r + 8 * half;          // D-matrix M = r + 8*half
            const int w = w0;                    // D-matrix N = lane&15
            if (o < 10 && w < 120) {
                float v = c[r] + convb[o];
                v = v > 0.f ? v : 0.f;
                atomicAdd(&acc[o * 2 + hh], v);  // ds_add_f32
            }
        }
    }
    __syncthreads();

    // ---------------- Phase 8: 20 -> 10 -> 2 head + softmax --------------
    if (tid == 0) {
        float h2[10];
        for (int q = 0; q < 10; ++q) {
            float s = fcn_b1[q];
            for (int j = 0; j < 20; ++j)
                s += (acc[j] * (1.f / 120.f)) * fcn_w1[q * 20 + j];
            h2[q] = 1.f / (1.f + expf(-s));
        }
        float l0 = fcn_b2[0], l1 = fcn_b2[1];
        for (int q = 0; q < 10; ++q) {
            l0 += h2[q] * fcn_w2[q];
            l1 += h2[q] * fcn_w2[10 + q];
        }
        const float mx = fmaxf(l0, l1);
        const float e0 = expf(l0 - mx), e1 = expf(l1 - mx);
        const float inv = 1.f / (e0 + e1);
        out[0] = e0 * inv;
        out[1] = e1 * inv;
    }
}

extern "C" void kernel_launch(void* const* d_in, const int* in_sizes, int n_in,
                              void* d_out, int out_size, void* d_ws, size_t ws_size,
                              hipStream_t stream) {
    (void)in_sizes; (void)n_in; (void)d_ws; (void)ws_size; (void)out_size;
    const float* x      = (const float*)d_in[0];
    const float* se_w1  = (const float*)d_in[1];
    const float* se_b1  = (const float*)d_in[2];
    const float* se_w2  = (const float*)d_in[3];
    const float* se_b2  = (const float*)d_in[4];
    const float* conv_w = (const float*)d_in[5];
    const float* conv_b = (const float*)d_in[6];
    const float* fcn_w1 = (const float*)d_in[7];
    const float* fcn_b1 = (const float*)d_in[8];
    const float* fcn_w2 = (const float*)d_in[9];
    const float* fcn_b2 = (const float*)d_in[10];
    float* out = (float*)d_out;

    eegnet_fused<<<1, 256, 0, stream>>>(x, se_w1, se_b1, se_w2, se_b2,
                                        conv_w, conv_b, fcn_w1, fcn_b1,
                                        fcn_w2, fcn_b2, out);
}